// GNN_16518444221123
// MI455X (gfx1250) — compile-verified
//
#include <hip/hip_runtime.h>
#include <hip/hip_bf16.h>

// GCN (4 layers, concat-initial-features, residual) for gfx1250.
// Dense per-node GEMMs use V_WMMA_F32_16X16X4_F32 (exact fp32 matrix op),
// edge aggregation uses global_atomic_add_f32. Memory-bound by edge traffic
// (~6.5 GB @ 23.3 TB/s); GEMM is ~23 GFLOP total, negligible.

typedef __attribute__((ext_vector_type(2))) float v2f;
typedef __attribute__((ext_vector_type(8))) float v8f;

#define EMB 128

// ---------------------------------------------------------------------------
// Zero fill (float4 wide)
// ---------------------------------------------------------------------------
__global__ void zero_kernel(float4* __restrict__ p, long n4) {
    long stride = (long)gridDim.x * blockDim.x;
    for (long i = (long)blockIdx.x * blockDim.x + threadIdx.x; i < n4; i += stride)
        p[i] = make_float4(0.0f, 0.0f, 0.0f, 0.0f);
}

// ---------------------------------------------------------------------------
// T[N,128] = op(X0) @ W[0:128,:]  (+ X1 @ W[128:256,:] when X1 != nullptr)
// op = relu when relu0 != 0. One wave per 16-row strip, full 128 columns.
// WMMA f32 16x16x4 VGPR layouts per CDNA5 ISA 7.12.2:
//   A 16x4 : lane L<16  -> row L,    K = k0+0, k0+1  (v2f)
//            lane L>=16 -> row L-16, K = k0+2, k0+3
//   B 4x16 : VGPR0 lanes0-15 K=k0+0, lanes16-31 K=k0+2
//            VGPR1 lanes0-15 K=k0+1, lanes16-31 K=k0+3
//   C/D    : VGPR i -> (M=i, N=lane) / (M=i+8, N=lane-16)
// ---------------------------------------------------------------------------
__global__ void __launch_bounds__(256) gemm_kernel(
    const float* __restrict__ X0, const float* __restrict__ X1,
    const float* __restrict__ W, float* __restrict__ T, int n, int relu0)
{
    const int lane  = threadIdx.x & 31;
    const int wave  = threadIdx.x >> 5;
    const int strip = blockIdx.x * 8 + wave;          // wave-uniform
    if (strip * 16 >= n) return;                      // whole wave exits -> EXEC stays all-1s
    const int row = strip * 16 + (lane & 15);
    const int kh  = (lane >> 4) << 1;                 // 0 or 2
    const int col = lane & 15;

    v8f acc[8];
#pragma unroll
    for (int t = 0; t < 8; ++t)
#pragma unroll
        for (int i = 0; i < 8; ++i) acc[t][i] = 0.0f;

    // ---- input 0 (optionally relu'd) against W rows [0,128) ----
    {
        const float* xrow = X0 + (size_t)row * EMB;
#pragma unroll 2
        for (int k0 = 0; k0 < 128; k0 += 4) {
            v2f a = *(const v2f*)(xrow + k0 + kh);    // 8B-aligned pair load
            if (relu0) { a.x = fmaxf(a.x, 0.0f); a.y = fmaxf(a.y, 0.0f); }
            const float* wk = W + (size_t)(k0 + kh) * EMB + col;
#pragma unroll
            for (int t = 0; t < 8; ++t) {
                v2f b;
                b.x = wk[t * 16];
                b.y = wk[t * 16 + EMB];
                acc[t] = __builtin_amdgcn_wmma_f32_16x16x4_f32(
                    false, a, false, b, (short)0, acc[t], false, false);
            }
        }
    }
    // ---- input 1 (raw x) against W rows [128,256) ----
    if (X1) {
        const float* xrow = X1 + (size_t)row * EMB;
        const float* Wlo  = W + 128 * EMB;
#pragma unroll 2
        for (int k0 = 0; k0 < 128; k0 += 4) {
            v2f a = *(const v2f*)(xrow + k0 + kh);
            const float* wk = Wlo + (size_t)(k0 + kh) * EMB + col;
#pragma unroll
            for (int t = 0; t < 8; ++t) {
                v2f b;
                b.x = wk[t * 16];
                b.y = wk[t * 16 + EMB];
                acc[t] = __builtin_amdgcn_wmma_f32_16x16x4_f32(
                    false, a, false, b, (short)0, acc[t], false, false);
            }
        }
    }

    // ---- store 16x128 strip ----
    const int rbase = strip * 16 + ((lane >> 4) << 3);
    float* tp = T + (size_t)rbase * EMB + col;
#pragma unroll
    for (int t = 0; t < 8; ++t)
#pragma unroll
        for (int i = 0; i < 8; ++i)
            tp[(size_t)i * EMB + t * 16] = acc[t][i];
}

// ---------------------------------------------------------------------------
// Edge scatter-sum: A[dst[e]] += T[src[e]]. One wave per edge, float4/lane.
// ---------------------------------------------------------------------------
__global__ void __launch_bounds__(256) scatter_kernel(
    const int* __restrict__ src, const int* __restrict__ dst,
    const float* __restrict__ T, float* __restrict__ A, int nE)
{
    const int e = blockIdx.x * 8 + (threadIdx.x >> 5);
    if (e >= nE) return;
    const int lane = threadIdx.x & 31;
    const int s = src[e];
    const int d = dst[e];
    float4 v = *((const float4*)(T + (size_t)s * EMB) + lane);
    float* ap = A + (size_t)d * EMB + lane * 4;
    atomicAdd(ap + 0, v.x);
    atomicAdd(ap + 1, v.y);
    atomicAdd(ap + 2, v.z);
    atomicAdd(ap + 3, v.w);
}

// ---------------------------------------------------------------------------
// h = (residual ? h : 0) + agg + bias[f]
// ---------------------------------------------------------------------------
__global__ void combine_kernel(float* __restrict__ h, const float* __restrict__ agg,
                               const float* __restrict__ bias, long total, int residual)
{
    long stride = (long)gridDim.x * blockDim.x;
    for (long idx = (long)blockIdx.x * blockDim.x + threadIdx.x; idx < total; idx += stride) {
        float v = agg[idx] + bias[idx & (EMB - 1)];
        if (residual) v += h[idx];
        h[idx] = v;
    }
}

// ---------------------------------------------------------------------------
// Root compaction (ordered nonzero of bool mask) in 3 passes + gather.
// ---------------------------------------------------------------------------
__global__ void mask_count_kernel(const unsigned char* __restrict__ mask, int n,
                                  int* __restrict__ blockCounts)
{
    __shared__ int sh[256];
    int i = blockIdx.x * 256 + threadIdx.x;
    sh[threadIdx.x] = (i < n && mask[i] != 0) ? 1 : 0;
    __syncthreads();
    for (int s = 128; s > 0; s >>= 1) {
        if (threadIdx.x < s) sh[threadIdx.x] += sh[threadIdx.x + s];
        __syncthreads();
    }
    if (threadIdx.x == 0) blockCounts[blockIdx.x] = sh[0];
}

__global__ void mask_scan_kernel(int* __restrict__ blockCounts, int nb) {
    if (threadIdx.x == 0 && blockIdx.x == 0) {
        int run = 0;
        for (int i = 0; i < nb; ++i) { int c = blockCounts[i]; blockCounts[i] = run; run += c; }
    }
}

__global__ void mask_write_kernel(const unsigned char* __restrict__ mask, int n,
                                  const int* __restrict__ blockOff,
                                  int* __restrict__ rootIdx, int maxRoots)
{
    __shared__ int sh[256];
    int i = blockIdx.x * 256 + threadIdx.x;
    int m = (i < n && mask[i] != 0) ? 1 : 0;
    sh[threadIdx.x] = m;
    __syncthreads();
    for (int off = 1; off < 256; off <<= 1) {   // inclusive Hillis-Steele
        int v = (threadIdx.x >= (unsigned)off) ? sh[threadIdx.x - off] : 0;
        __syncthreads();
        sh[threadIdx.x] += v;
        __syncthreads();
    }
    if (m) {
        int rank = blockOff[blockIdx.x] + sh[threadIdx.x] - 1;
        if (rank < maxRoots) rootIdx[rank] = i;
    }
}

__global__ void gather_kernel(const float* __restrict__ h, const int* __restrict__ rootIdx,
                              float* __restrict__ out, int nroots)
{
    int idx = blockIdx.x * blockDim.x + threadIdx.x;
    if (idx >= nroots * EMB) return;
    int r = idx >> 7, f = idx & (EMB - 1);
    out[idx] = h[(size_t)rootIdx[r] * EMB + f];
}

// ---------------------------------------------------------------------------
extern "C" void kernel_launch(void* const* d_in, const int* in_sizes, int n_in,
                              void* d_out, int out_size, void* d_ws, size_t ws_size,
                              hipStream_t stream)
{
    const float*         x    = (const float*)d_in[0];
    const int*           ei   = (const int*)d_in[1];
    const unsigned char* mask = (const unsigned char*)d_in[2];
    // d_in[3] = num_roots scalar (unused; derived from out_size)
    const float* W[4] = { (const float*)d_in[4], (const float*)d_in[6],
                          (const float*)d_in[8], (const float*)d_in[10] };
    const float* B[4] = { (const float*)d_in[5], (const float*)d_in[7],
                          (const float*)d_in[9], (const float*)d_in[11] };

    const int n = in_sizes[0] / EMB;
    const int E = in_sizes[1] / 2;
    const int* src = ei;
    const int* dst = ei + E;
    const int nroots = out_size / EMB;

    const size_t nf = (size_t)n * EMB;
    float* tbuf = (float*)d_ws;        // per-node transformed features
    float* abuf = tbuf + nf;           // per-node aggregated messages
    float* hbuf = abuf + nf;           // running hidden state
    const int nb = (n + 255) / 256;
    int* blockOff = (int*)(hbuf + nf);
    int* rootIdx  = blockOff + nb;

    const dim3 blk(256);
    const int gemmBlocks = (((n + 15) / 16) + 7) / 8;
    const int scatBlocks = (E + 7) / 8;

    auto layer = [&](const float* X0, const float* X1, const float* Wk,
                     const float* bk, int relu0, int residual) {
        zero_kernel<<<dim3(2048), blk, 0, stream>>>((float4*)abuf, (long)(nf / 4));
        gemm_kernel<<<dim3(gemmBlocks), blk, 0, stream>>>(X0, X1, Wk, tbuf, n, relu0);
        scatter_kernel<<<dim3(scatBlocks), blk, 0, stream>>>(src, dst, tbuf, abuf, E);
        combine_kernel<<<dim3(2048), blk, 0, stream>>>(hbuf, abuf, bk, (long)nf, residual);
    };

    layer(x,    nullptr, W[0], B[0], 0, 0);   // h = GCN0(x)
    layer(hbuf, x,       W[1], B[1], 1, 1);   // h += GCN1(concat(relu(h), x))
    layer(hbuf, x,       W[2], B[2], 1, 1);
    layer(hbuf, x,       W[3], B[3], 1, 1);

    mask_count_kernel<<<dim3(nb), blk, 0, stream>>>(mask, n, blockOff);
    mask_scan_kernel<<<dim3(1), dim3(32), 0, stream>>>(blockOff, nb);
    mask_write_kernel<<<dim3(nb), blk, 0, stream>>>(mask, n, blockOff, rootIdx, nroots);
    gather_kernel<<<dim3((nroots * EMB + 255) / 256), blk, 0, stream>>>(hbuf, rootIdx,
                                                                        (float*)d_out, nroots);
}